// GAT_13975823581722
// MI455X (gfx1250) — compile-verified
//
#include <hip/hip_runtime.h>
#include <math.h>

#define N_NODES 50000
#define N_EDGES 800000
#define HEADS 3
#define HID 64
#define NCOL (HEADS * HID)   // 192
#define NUM_CLASSES 8
#define ATTN_SLOPE 0.2f
#define ACT_SLOPE 0.01f

typedef __attribute__((ext_vector_type(2))) float v2f;
typedef __attribute__((ext_vector_type(8))) float v8f;

// ---------------- utility ----------------

__global__ void fill_f32(float* __restrict__ p, float v, int n) {
    int t = blockIdx.x * blockDim.x + threadIdx.x;
    if (t < n) p[t] = v;
}

__device__ __forceinline__ float atomicMaxFloat(float* addr, float value) {
    // works for any sign: positive floats compare as ints, negative as reversed unsigned
    if (value >= 0.0f)
        return __int_as_float(atomicMax((int*)addr, __float_as_int(value)));
    else
        return __uint_as_float(atomicMin((unsigned int*)addr, __float_as_uint(value)));
}

// ---------------- node linear: feat = H @ W + b via V_WMMA_F32_16X16X4_F32 ----------------
// grid = (M/16, NCOL/16), block = 32 (one wave per 16x16 output tile)

__global__ void node_linear_wmma(const float* __restrict__ H,
                                 const float* __restrict__ W,
                                 const float* __restrict__ bias,
                                 float* __restrict__ feat,
                                 int K) {
    const int tM   = blockIdx.x;
    const int tN   = blockIdx.y;
    const int lane = threadIdx.x;      // 0..31, wave32
    const int half = lane >> 4;        // 0: K pair {0,1}, 1: K pair {2,3}
    const int l15  = lane & 15;

    const int row = tM * 16 + l15;     // A row for this lane
    const int col = tN * 16 + l15;     // B/C column for this lane

    v8f acc = {};
    const float* __restrict__ hrow = H + (size_t)row * K;

    for (int k = 0; k < K; k += 4) {
        // A fragment 16x4: lane holds K = k + 2*half + {0,1} of its row
        v2f a = *(const v2f*)(hrow + k + 2 * half);
        // B fragment 4x16: lane holds rows K = k + 2*half + {0,1} at its column
        v2f b;
        b.x = W[(size_t)(k + 2 * half) * NCOL + col];
        b.y = W[(size_t)(k + 2 * half + 1) * NCOL + col];
        acc = __builtin_amdgcn_wmma_f32_16x16x4_f32(
            /*neg_a=*/false, a, /*neg_b=*/false, b,
            /*c_mod=*/(short)0, acc, /*reuse_a=*/false, /*reuse_b=*/false);
    }

    // C/D layout: VGPR r -> M = r + 8*half (lanes 0-15: M 0..7, lanes 16-31: M 8..15)
    const float bcol = bias[col];
    const int orow = tM * 16 + half * 8;
#pragma unroll
    for (int r = 0; r < 8; ++r)
        feat[(size_t)(orow + r) * NCOL + col] = acc[r] + bcol;
}

// ---------------- el/er: per (node, head) dot of feat row with al/ar ----------------

__global__ void attn_coeff(const float* __restrict__ feat,
                           const float* __restrict__ al,
                           const float* __restrict__ ar,
                           float* __restrict__ el,
                           float* __restrict__ er) {
    int t = blockIdx.x * blockDim.x + threadIdx.x;   // over N*HEADS
    if (t >= N_NODES * HEADS) return;
    int n = t / HEADS, h = t % HEADS;
    const float* __restrict__ f  = feat + (size_t)n * NCOL + h * HID;
    const float* __restrict__ a1 = al + h * HID;
    const float* __restrict__ a2 = ar + h * HID;
    float s1 = 0.f, s2 = 0.f;
#pragma unroll 4
    for (int i = 0; i < HID; ++i) {
        float v = f[i];
        s1 += v * a1[i];
        s2 += v * a2[i];
    }
    el[t] = s1;
    er[t] = s2;
}

// ---------------- edge pass 1: segment max of leaky_relu(el[src]+er[dst]) ----------------

__global__ void edge_max(const int* __restrict__ src, const int* __restrict__ dst,
                         const float* __restrict__ el, const float* __restrict__ er,
                         float* __restrict__ maxb) {
    int e = blockIdx.x * blockDim.x + threadIdx.x;
    if (e >= N_EDGES) return;
    int s = src[e], d = dst[e];
#pragma unroll
    for (int h = 0; h < HEADS; ++h) {
        float x = el[s * HEADS + h] + er[d * HEADS + h];
        x = x > 0.f ? x : ATTN_SLOPE * x;
        atomicMaxFloat(&maxb[d * HEADS + h], x);
    }
}

// ---------------- edge pass 2: ex = exp(e - m[dst]); denom = segsum(ex) ----------------

__global__ void edge_exp(const int* __restrict__ src, const int* __restrict__ dst,
                         const float* __restrict__ el, const float* __restrict__ er,
                         const float* __restrict__ maxb,
                         float* __restrict__ den, float* __restrict__ exb) {
    int e = blockIdx.x * blockDim.x + threadIdx.x;
    if (e >= N_EDGES) return;
    int s = src[e], d = dst[e];
#pragma unroll
    for (int h = 0; h < HEADS; ++h) {
        float x = el[s * HEADS + h] + er[d * HEADS + h];
        x = x > 0.f ? x : ATTN_SLOPE * x;
        float ex = __expf(x - maxb[d * HEADS + h]);
        exb[(size_t)e * HEADS + h] = ex;
        atomicAdd(&den[d * HEADS + h], ex);
    }
}

// ---------------- edge pass 3: agg[dst] += alpha * feat[src]  (wave per edge-head) ----------------

__global__ void edge_scatter(const int* __restrict__ src, const int* __restrict__ dst,
                             const float* __restrict__ exb, const float* __restrict__ den,
                             const float* __restrict__ feat, float* __restrict__ agg) {
    int gid  = blockIdx.x * blockDim.x + threadIdx.x;
    int widx = gid >> 5;         // one wave per (edge, head)
    int lane = gid & 31;
    if (widx >= N_EDGES * HEADS) return;
    int e = widx / HEADS, h = widx % HEADS;
    int s = src[e], d = dst[e];
    float alpha = exb[(size_t)e * HEADS + h] / den[d * HEADS + h];
    const float* __restrict__ fsrc = feat + (size_t)s * NCOL + h * HID;
    float* __restrict__ o = agg + (size_t)d * NCOL + h * HID;
    int f = lane * 2;            // 64 floats / 32 lanes = 2 each
    atomicAdd(&o[f],     alpha * fsrc[f]);
    atomicAdd(&o[f + 1], alpha * fsrc[f + 1]);
}

// ---------------- post: h = mean_h leaky_relu(agg + b, 0.01) ----------------

__global__ void post_mean(const float* __restrict__ agg, const float* __restrict__ bias,
                          float* __restrict__ hnext) {
    int t = blockIdx.x * blockDim.x + threadIdx.x;   // over N*HID
    if (t >= N_NODES * HID) return;
    int n = t >> 6, f = t & 63;
    float s = 0.f;
#pragma unroll
    for (int h = 0; h < HEADS; ++h) {
        float v = agg[(size_t)n * NCOL + h * HID + f] + bias[h * HID + f];
        s += v > 0.f ? v : ACT_SLOPE * v;
    }
    hnext[t] = s * (1.0f / HEADS);
}

// ---------------- final linear: out = h @ Wo + bo ----------------

__global__ void final_linear(const float* __restrict__ h, const float* __restrict__ Wo,
                             const float* __restrict__ bo, float* __restrict__ out) {
    int t = blockIdx.x * blockDim.x + threadIdx.x;   // over N*NUM_CLASSES
    if (t >= N_NODES * NUM_CLASSES) return;
    int n = t >> 3, c = t & 7;
    const float* __restrict__ hr = h + (size_t)n * HID;
    float s = bo[c];
#pragma unroll 8
    for (int k = 0; k < HID; ++k) s += hr[k] * Wo[k * NUM_CLASSES + c];
    out[t] = s;
}

// ---------------- launch ----------------

extern "C" void kernel_launch(void* const* d_in, const int* in_sizes, int n_in,
                              void* d_out, int out_size, void* d_ws, size_t ws_size,
                              hipStream_t stream) {
    const float* x   = (const float*)d_in[0];
    const int*   src = (const int*)d_in[1];
    const int*   dst = (const int*)d_in[2];
    const float *W[5], *b[5], *al[5], *ar[5];
    for (int l = 0; l < 5; ++l) {
        W[l]  = (const float*)d_in[3 + l * 4];
        b[l]  = (const float*)d_in[4 + l * 4];
        al[l] = (const float*)d_in[5 + l * 4];
        ar[l] = (const float*)d_in[6 + l * 4];
    }
    const float* Wo = (const float*)d_in[23];
    const float* bo = (const float*)d_in[24];
    float* out = (float*)d_out;

    // workspace carve-out
    char* ws = (char*)d_ws;
    size_t off = 0;
    auto carve = [&](size_t bytes) -> float* {
        float* p = (float*)(ws + off);
        off = (off + bytes + 255) & ~(size_t)255;
        return p;
    };
    float* feat = carve((size_t)N_NODES * NCOL * 4);       // 38.4 MB
    float* agg  = carve((size_t)N_NODES * NCOL * 4);       // 38.4 MB
    float* el   = carve((size_t)N_NODES * HEADS * 4);
    float* er   = carve((size_t)N_NODES * HEADS * 4);
    float* maxb = carve((size_t)N_NODES * HEADS * 4);
    float* den  = carve((size_t)N_NODES * HEADS * 4);
    float* exb  = carve((size_t)N_EDGES * HEADS * 4);      // 9.6 MB
    float* hbuf = carve((size_t)N_NODES * HID * 4);        // 12.8 MB

    const int NH   = N_NODES * HEADS;
    const int NF   = N_NODES * HID;
    const int NAGG = N_NODES * NCOL;

    const float* hin = x;
    int K = 128;
    for (int l = 0; l < 5; ++l) {
        dim3 g(N_NODES / 16, NCOL / 16);                   // 3125 x 12, exact
        node_linear_wmma<<<g, 32, 0, stream>>>(hin, W[l], b[l], feat, K);

        attn_coeff<<<(NH + 255) / 256, 256, 0, stream>>>(feat, al[l], ar[l], el, er);

        fill_f32<<<(NH + 255) / 256, 256, 0, stream>>>(maxb, -__builtin_inff(), NH);
        fill_f32<<<(NH + 255) / 256, 256, 0, stream>>>(den, 0.f, NH);
        fill_f32<<<(NAGG + 255) / 256, 256, 0, stream>>>(agg, 0.f, NAGG);

        edge_max<<<(N_EDGES + 255) / 256, 256, 0, stream>>>(src, dst, el, er, maxb);
        edge_exp<<<(N_EDGES + 255) / 256, 256, 0, stream>>>(src, dst, el, er, maxb, den, exb);

        long long scatter_threads = (long long)N_EDGES * HEADS * 32;   // 76.8M
        edge_scatter<<<(int)((scatter_threads + 255) / 256), 256, 0, stream>>>(
            src, dst, exb, den, feat, agg);

        post_mean<<<(NF + 255) / 256, 256, 0, stream>>>(agg, b[l], hbuf);

        hin = hbuf;
        K = HID;
    }

    final_linear<<<(N_NODES * NUM_CLASSES + 255) / 256, 256, 0, stream>>>(hbuf, Wo, bo, out);
}